// S4DKernel_55362128445894
// MI455X (gfx1250) — compile-verified
//
#include <hip/hip_runtime.h>
#include <math.h>

typedef float v2f __attribute__((ext_vector_type(2)));
typedef float v8f __attribute__((ext_vector_type(8)));

#define H_DIM 1024
#define N2 32
#define L_LEN 4096
#define T_TILE 64       // inner time index t (columns, N of GEMM)
#define LO_TILE 64      // outer index lo (rows, M of GEMM); l = lo*64 + t
#define KDIM 64         // 2*N2 (real/imag interleaved inner dim)
#define STRIDE 66       // padded LDS row stride (floats); even -> 8B-aligned v2f

__global__ __launch_bounds__(128)
void s4d_vandermonde_wmma(const float* __restrict__ log_dt,
                          const float* __restrict__ A_imag,
                          const float* __restrict__ log_A_real,
                          const float* __restrict__ Cm,
                          float* __restrict__ out)
{
    __shared__ float sA[LO_TILE * STRIDE]; // sA[lo][k]: k=2n -> u_r, k=2n+1 -> -u_i
    __shared__ float sB[T_TILE * STRIDE];  // sB[t][k]:  k=2n -> v_r, k=2n+1 ->  v_i

    const int h   = blockIdx.x;
    const int tid = threadIdx.x;

    const float dt = expf(log_dt[h]);

    // ---- Phase 1a: B operand, v[n][t] = exp(dA_n * t), stored transposed ----
    for (int idx = tid; idx < N2 * T_TILE; idx += 128) {
        const int n = idx >> 6;
        const int t = idx & 63;
        const float ar  = -expf(log_A_real[h * N2 + n]);
        const float ai  = A_imag[h * N2 + n];
        const float dAr = ar * dt;
        const float dAi = ai * dt;
        const float e = expf(dAr * (float)t);
        float s, c;
        sincosf(dAi * (float)t, &s, &c);
        sB[t * STRIDE + 2 * n + 0] = e * c;
        sB[t * STRIDE + 2 * n + 1] = e * s;
    }

    // ---- Phase 1b: A operand, u[n][lo] = c'_n * exp(dA_n * 64 * lo) ----
    for (int idx = tid; idx < N2 * LO_TILE; idx += 128) {
        const int n  = idx >> 6;
        const int lo = idx & 63;
        const float ar  = -expf(log_A_real[h * N2 + n]);
        const float ai  = A_imag[h * N2 + n];
        const float dAr = ar * dt;
        const float dAi = ai * dt;
        // c' = 2 * C * (exp(dA) - 1) / A   (complex)
        float sd, cd;
        sincosf(dAi, &sd, &cd);
        const float edr = expf(dAr);
        const float nr  = edr * cd - 1.0f;
        const float ni  = edr * sd;
        const float inv = 1.0f / (ar * ar + ai * ai);
        const float fr  = (nr * ar + ni * ai) * inv;   // (num * conj(A)) / |A|^2
        const float fi  = (ni * ar - nr * ai) * inv;
        const float Cr  = Cm[(h * N2 + n) * 2 + 0];
        const float Ci  = Cm[(h * N2 + n) * 2 + 1];
        const float cpr = 2.0f * (Cr * fr - Ci * fi);
        const float cpi = 2.0f * (Cr * fi + Ci * fr);
        // p = exp(dA * (64*lo))
        const float ll = (float)(T_TILE * lo);
        const float ep = expf(dAr * ll);
        float sp, cp;
        sincosf(dAi * ll, &sp, &cp);
        const float pr = ep * cp;
        const float pi = ep * sp;
        sA[lo * STRIDE + 2 * n + 0] =  (cpr * pr - cpi * pi);  //  u_r
        sA[lo * STRIDE + 2 * n + 1] = -(cpr * pi + cpi * pr);  // -u_i
    }

    __syncthreads();

    // ---- Phase 2: per-wave 16x64 strip via V_WMMA_F32_16X16X4_F32 ----
    const int lane = tid & 31;
    const int wave = tid >> 5;
    const int m0   = wave * 16;           // M-tile base (lo rows)
    const int row  = lane & 15;           // M index (A) / N index (B) within tile
    const int kb   = (lane >> 4) << 1;    // K sub-offset: lanes 0-15 -> {K0,K1}, 16-31 -> {K2,K3}

    v8f acc0 = {0.f,0.f,0.f,0.f,0.f,0.f,0.f,0.f};
    v8f acc1 = {0.f,0.f,0.f,0.f,0.f,0.f,0.f,0.f};
    v8f acc2 = {0.f,0.f,0.f,0.f,0.f,0.f,0.f,0.f};
    v8f acc3 = {0.f,0.f,0.f,0.f,0.f,0.f,0.f,0.f};

    #pragma unroll
    for (int kk = 0; kk < 16; ++kk) {
        const int k0 = kk * 4 + kb;
        const v2f a  = *(const v2f*)&sA[(m0 + row) * STRIDE + k0];
        const v2f b0 = *(const v2f*)&sB[(0 * 16 + row) * STRIDE + k0];
        const v2f b1 = *(const v2f*)&sB[(1 * 16 + row) * STRIDE + k0];
        const v2f b2 = *(const v2f*)&sB[(2 * 16 + row) * STRIDE + k0];
        const v2f b3 = *(const v2f*)&sB[(3 * 16 + row) * STRIDE + k0];
        acc0 = __builtin_amdgcn_wmma_f32_16x16x4_f32(false, a, false, b0, (short)0, acc0, false, false);
        acc1 = __builtin_amdgcn_wmma_f32_16x16x4_f32(false, a, false, b1, (short)0, acc1, false, false);
        acc2 = __builtin_amdgcn_wmma_f32_16x16x4_f32(false, a, false, b2, (short)0, acc2, false, false);
        acc3 = __builtin_amdgcn_wmma_f32_16x16x4_f32(false, a, false, b3, (short)0, acc3, false, false);
    }

    // ---- Phase 3: write K[h, lo*64 + t] from C/D tile layout ----
    float* __restrict__ outh = out + (size_t)h * L_LEN;
    const int radd = (lane >> 4) << 3;    // lanes 16-31 hold M = r + 8
    #pragma unroll
    for (int r = 0; r < 8; ++r) {
        const int lo = m0 + r + radd;
        outh[lo * 64 + 0 * 16 + row] = acc0[r];
        outh[lo * 64 + 1 * 16 + row] = acc1[r];
        outh[lo * 64 + 2 * 16 + row] = acc2[r];
        outh[lo * 64 + 3 * 16 + row] = acc3[r];
    }
}

extern "C" void kernel_launch(void* const* d_in, const int* in_sizes, int n_in,
                              void* d_out, int out_size, void* d_ws, size_t ws_size,
                              hipStream_t stream) {
    const float* log_dt     = (const float*)d_in[0];
    const float* A_imag     = (const float*)d_in[1];
    const float* log_A_real = (const float*)d_in[2];
    const float* C          = (const float*)d_in[3];
    float* out = (float*)d_out;
    s4d_vandermonde_wmma<<<H_DIM, 128, 0, stream>>>(log_dt, A_imag, log_A_real, C, out);
}